// SSIMLoss_68393059222027
// MI455X (gfx1250) — compile-verified
//
#include <hip/hip_runtime.h>

// SSIM loss on MI455X (gfx1250, wave32).
// - Phase 1: tile staging via Tensor Data Mover (tensor_load_to_lds, TENSORcnt)
// - Phase 2: horizontal 7-sums of 5 channels (sliding window, VALU)
// - Phase 3: vertical 7-sums on matrix pipes: V_WMMA_F32_16X16X4_F32 against a
//            banded 0/1 matrix; B operands as ds_load_b64 (column-major hsum)
// - Phase 4: SSIM map in registers; deterministic reduction.

#define T_FRAMES 128
#define IMG_H 512
#define IMG_W 512
#define OH 506
#define OW 506

#define TM 32            // output rows per block
#define TN 48            // output cols per block
#define IN_ROWS 38       // TM + 6
#define IN_COLS 54       // TN + 6 (also LDS row stride; TDM writes contiguous)
#define HRS 40           // hs row stride (38 data rows + 2 zero pad, even->b64 ok)
#define NWAVES 6
#define BLOCK 192        // 6 wave32 waves
#define GX 11            // ceil(506/48)
#define GY 16            // ceil(506/32)
#define NBLK (GX * GY * T_FRAMES)

typedef float v2f __attribute__((ext_vector_type(2)));
typedef float v8f __attribute__((ext_vector_type(8)));
typedef unsigned int v4u __attribute__((ext_vector_type(4)));
typedef int v4i __attribute__((ext_vector_type(4)));
typedef int v8i __attribute__((ext_vector_type(8)));

#if defined(__gfx1250__) && __has_builtin(__builtin_amdgcn_tensor_load_to_lds) && \
    __has_builtin(__builtin_amdgcn_s_wait_tensorcnt)
#define HAVE_TDM 1
#else
#define HAVE_TDM 0
#endif

#if HAVE_TDM
// Issue one 2D TDM load: tile of tile_d1 x tile_d0 f32 from a row-major tensor
// whose remaining extent from the tile origin is td1 x td0 (OOB reads -> 0).
__device__ __forceinline__ void tdm_load_2d_f32(unsigned lds_byte_addr,
                                                const float* tile_origin,
                                                unsigned td0, unsigned td1,
                                                unsigned tile_d0, unsigned tile_d1,
                                                unsigned row_stride) {
  const unsigned long long ga = (unsigned long long)(uintptr_t)tile_origin;
  v4u g0;
  g0.x = 1u;                                    // count=1, user descriptor
  g0.y = lds_byte_addr;                         // lds_addr
  g0.z = (unsigned)ga;                          // global_addr[31:0]
  g0.w = (unsigned)((ga >> 32) & 0x1FFFFFFu) | (2u << 30);  // addr[56:32] | type=2
  v8i g1;
  g1[0] = (int)(2u << 16);                      // workgroup_mask=0, data_size=2 (4B)
  g1[1] = (int)((td0 & 0xFFFFu) << 16);         // tensor_dim0[15:0]
  g1[2] = (int)(((td0 >> 16) & 0xFFFFu) | ((td1 & 0xFFFFu) << 16));
  g1[3] = (int)(((td1 >> 16) & 0xFFFFu) | (tile_d0 << 16));  // tile_dim0
  g1[4] = (int)tile_d1;                         // tile_dim1, tile_dim2=0
  g1[5] = (int)row_stride;                      // tensor_dim0_stride[31:0]
  g1[6] = 0;                                    // stride hi, dim1_stride lo
  g1[7] = 0;
  v4i z4 = {};
#if __clang_major__ >= 23
  v8i z8 = {};
  __builtin_amdgcn_tensor_load_to_lds(g0, g1, z4, z4, z8, 0);
#else
  __builtin_amdgcn_tensor_load_to_lds(g0, g1, z4, z4, 0);
#endif
}
#endif

__global__ __launch_bounds__(BLOCK) void ssim_tile_kernel(
    const float* __restrict__ x, const float* __restrict__ y,
    float* __restrict__ partial) {
  __shared__ float sx[IN_ROWS][IN_COLS];
  __shared__ float sy[IN_ROWS][IN_COLS];
  __shared__ float hs[5][TN][HRS];   // column-major: [channel][out-col][in-row]
  __shared__ float wsum[NWAVES];

  const int tid = threadIdx.x;
  const int wv = tid >> 5;
  const int lane = tid & 31;
  const int r0 = blockIdx.y * TM;
  const int c0 = blockIdx.x * TN;
  const size_t frame = (size_t)blockIdx.z * (IMG_H * IMG_W);

  // Zero the two hs pad rows (read when the band matrix is 0; keeps 0*0=0).
  for (int idx = tid; idx < 5 * TN; idx += BLOCK) {
    const int ch = idx / TN, cc = idx % TN;
    hs[ch][cc][IN_ROWS] = 0.0f;
    hs[ch][cc][IN_ROWS + 1] = 0.0f;
  }

  // ---- Phase 1: stage x,y tiles into LDS ----
#if HAVE_TDM
  if (wv == 0) {
    const float* xo = x + frame + (size_t)r0 * IMG_W + c0;
    const float* yo = y + frame + (size_t)r0 * IMG_W + c0;
    const unsigned td0 = (unsigned)(IMG_W - c0);   // remaining cols: OOB -> 0
    const unsigned td1 = (unsigned)(IMG_H - r0);   // remaining rows: OOB -> 0
    tdm_load_2d_f32((unsigned)(uintptr_t)&sx[0][0], xo, td0, td1,
                    IN_COLS, IN_ROWS, IMG_W);
    tdm_load_2d_f32((unsigned)(uintptr_t)&sy[0][0], yo, td0, td1,
                    IN_COLS, IN_ROWS, IMG_W);
    __builtin_amdgcn_s_wait_tensorcnt(0);
  }
#else
  for (int idx = tid; idx < IN_ROWS * IN_COLS; idx += BLOCK) {
    const int rr = idx / IN_COLS, cc = idx % IN_COLS;
    const int gr = r0 + rr, gc = c0 + cc;
    const bool ok = (gr < IMG_H) && (gc < IMG_W);
    const size_t off = frame + (size_t)gr * IMG_W + gc;
    sx[rr][cc] = ok ? x[off] : 0.0f;
    sy[rr][cc] = ok ? y[off] : 0.0f;
  }
#endif
  __syncthreads();

  // ---- Phase 2: horizontal 7-sums, sliding window, column-major store ----
  // 38 rows x 4 segments of 12 output columns = 152 tasks.
  for (int t = tid; t < IN_ROWS * 4; t += BLOCK) {
    const int rr = t >> 2;
    const int cs = (t & 3) * 12;
    float s0 = 0.f, s1 = 0.f, s2 = 0.f, s3 = 0.f, s4 = 0.f;
#pragma unroll
    for (int j = 0; j < 7; ++j) {
      const float xv = sx[rr][cs + j];
      const float yv = sy[rr][cs + j];
      s0 += xv; s1 += yv;
      s2 = fmaf(xv, xv, s2);
      s3 = fmaf(yv, yv, s3);
      s4 = fmaf(xv, yv, s4);
    }
    hs[0][cs][rr] = s0; hs[1][cs][rr] = s1; hs[2][cs][rr] = s2;
    hs[3][cs][rr] = s3; hs[4][cs][rr] = s4;
#pragma unroll
    for (int c = 1; c < 12; ++c) {
      const float ox = sx[rr][cs + c - 1], oy = sy[rr][cs + c - 1];
      const float nx = sx[rr][cs + c + 6], ny = sy[rr][cs + c + 6];
      s0 += nx - ox;
      s1 += ny - oy;
      s2 += fmaf(nx, nx, -ox * ox);
      s3 += fmaf(ny, ny, -oy * oy);
      s4 += fmaf(nx, ny, -ox * oy);
      hs[0][cs + c][rr] = s0; hs[1][cs + c][rr] = s1; hs[2][cs + c][rr] = s2;
      hs[3][cs + c][rr] = s3; hs[4][cs + c][rr] = s4;
    }
  }
  __syncthreads();

  // ---- Phase 3: vertical 7-sums via WMMA: Out = Band(16x24) x Hsum(24x16) ----
  // A (16x4 f32 layout: lanes 0-15 -> K0,K1; lanes 16-31 -> K2,K3) is the 0/1
  // band A[m][k] = (m <= k <= m+6). B pairs (K0,K1) are contiguous in
  // column-major hs -> single ds_load_b64 each.
  const int tr = wv / 3, tc = wv % 3;       // 2 x 3 sub-tiles of 16x16
  const int rowbase = tr * 16;
  const int colbase = tc * 16;
  const int m = lane & 15;
  const int khalf = lane >> 4;
  const int colL = colbase + m;             // B: N index == lane&15

  v8f acc0 = {}, acc1 = {}, acc2 = {}, acc3 = {}, acc4 = {};
#pragma unroll
  for (int chunk = 0; chunk < 6; ++chunk) {
    const int k0 = chunk * 4 + khalf * 2;   // even -> 8B aligned pair
    const int k1 = k0 + 1;
    v2f a;
    a.x = (k0 >= m && k0 <= m + 6) ? 1.0f : 0.0f;
    a.y = (k1 >= m && k1 <= m + 6) ? 1.0f : 0.0f;
    const int rw = rowbase + k0;            // <= 38; pad rows are zero
    v2f b;
    b = *(const v2f*)&hs[0][colL][rw];
    acc0 = __builtin_amdgcn_wmma_f32_16x16x4_f32(false, a, false, b, (short)0, acc0, false, false);
    b = *(const v2f*)&hs[1][colL][rw];
    acc1 = __builtin_amdgcn_wmma_f32_16x16x4_f32(false, a, false, b, (short)0, acc1, false, false);
    b = *(const v2f*)&hs[2][colL][rw];
    acc2 = __builtin_amdgcn_wmma_f32_16x16x4_f32(false, a, false, b, (short)0, acc2, false, false);
    b = *(const v2f*)&hs[3][colL][rw];
    acc3 = __builtin_amdgcn_wmma_f32_16x16x4_f32(false, a, false, b, (short)0, acc3, false, false);
    b = *(const v2f*)&hs[4][colL][rw];
    acc4 = __builtin_amdgcn_wmma_f32_16x16x4_f32(false, a, false, b, (short)0, acc4, false, false);
  }

  // ---- Phase 4: SSIM map in registers (C/D layouts align across channels) ----
  const float inv49 = 1.0f / 49.0f;
  const float covn = 49.0f / 48.0f;
  const float C1 = 4.0e-4f;   // (0.01*2)^2
  const float C2 = 36.0e-4f;  // (0.03*2)^2
  float local = 0.0f;
#pragma unroll
  for (int e = 0; e < 8; ++e) {
    const int orow = r0 + rowbase + e + 8 * khalf;  // C: lanes>=16 hold M=e+8
    const int ocol = c0 + colbase + m;
    const float ux = acc0[e] * inv49;
    const float uy = acc1[e] * inv49;
    const float uxx = acc2[e] * inv49;
    const float uyy = acc3[e] * inv49;
    const float uxy = acc4[e] * inv49;
    const float vx = covn * (uxx - ux * ux);
    const float vy = covn * (uyy - uy * uy);
    const float vxy = covn * (uxy - ux * uy);
    const float A1 = 2.0f * ux * uy + C1;
    const float A2 = 2.0f * vxy + C2;
    const float B1 = ux * ux + uy * uy + C1;
    const float B2 = vx + vy + C2;
    const float S = (A1 * A2) / (B1 * B2);
    local += (orow < OH && ocol < OW) ? S : 0.0f;
  }

  // ---- Phase 5: deterministic block reduction -> per-block partial ----
#pragma unroll
  for (int off = 16; off > 0; off >>= 1) local += __shfl_xor(local, off, 32);
  if (lane == 0) wsum[wv] = local;
  __syncthreads();
  if (tid == 0) {
    float t = 0.0f;
#pragma unroll
    for (int w = 0; w < NWAVES; ++w) t += wsum[w];
    const int bid = ((int)blockIdx.z * GY + (int)blockIdx.y) * GX + (int)blockIdx.x;
    partial[bid] = t;
  }
}

__global__ __launch_bounds__(256) void ssim_finalize_kernel(
    const float* __restrict__ partial, float* __restrict__ out) {
  __shared__ float s[8];
  float local = 0.0f;
  for (int i = threadIdx.x; i < NBLK; i += 256) local += partial[i];
#pragma unroll
  for (int off = 16; off > 0; off >>= 1) local += __shfl_xor(local, off, 32);
  if ((threadIdx.x & 31) == 0) s[threadIdx.x >> 5] = local;
  __syncthreads();
  if (threadIdx.x == 0) {
    float t = 0.0f;
#pragma unroll
    for (int w = 0; w < 8; ++w) t += s[w];
    // mean over frames of (1 - mean(S)) == 1 - total / (T*OH*OW)
    out[0] = 1.0f - t * (1.0f / (128.0f * 506.0f * 506.0f));
  }
}

extern "C" void kernel_launch(void* const* d_in, const int* in_sizes, int n_in,
                              void* d_out, int out_size, void* d_ws, size_t ws_size,
                              hipStream_t stream) {
  const float* x = (const float*)d_in[0];
  const float* y = (const float*)d_in[1];
  float* partial = (float*)d_ws;  // NBLK floats (= 88 KB)
  dim3 grid(GX, GY, T_FRAMES);
  ssim_tile_kernel<<<grid, dim3(BLOCK), 0, stream>>>(x, y, partial);
  ssim_finalize_kernel<<<1, 256, 0, stream>>>(partial, (float*)d_out);
}